// ChaosNetModel_81028853006451
// MI455X (gfx1250) — compile-verified
//
#include <hip/hip_runtime.h>
#include <math.h>

// ---------------------------------------------------------------------------
// ChaosFEX feature extraction + linear head, CDNA5 (gfx1250, wave32).
//
//   traj[k]   : shared skew-tent trajectory, length T=10000 (sequential)
//   per elem  : k_fire = first k with |traj[k]-x| < eps
//               tt = k_fire+1 (or T), energy = prefE[k_fire], cnt = prefC[k_fire]
//               p = cnt/tt, ent = -(p log2 p + (1-p) log2 (1-p))
//   out       : feats[2048,128] @ W^T[128,2] + b   (fp32)
//
// GEMM uses V_WMMA_F32_16X16X4_F32 (fp32-native CDNA5 WMMA), N padded 2->16,
// with a zero-padded B tile staged in LDS so the WMMA loop has no exec-mask
// divergence (WMMA requires EXEC all-ones anyway).
// ---------------------------------------------------------------------------

#define TRAJ_LEN 10000
#define CHUNK    2048          // 8 KB LDS per block for trajectory staging
#define NELEM    (2048 * 32)   // N * F

typedef __attribute__((ext_vector_type(2))) float v2f;
typedef __attribute__((ext_vector_type(8))) float v8f;

// ---- Kernel 1: sequential trajectory + prefix sums (inherently serial) ----
__global__ void chaos_traj_kernel(const float* __restrict__ ic,
                                  const float* __restrict__ thrp,
                                  float* __restrict__ traj,
                                  float* __restrict__ prefE,
                                  float* __restrict__ prefC) {
  if (blockIdx.x == 0 && threadIdx.x == 0) {
    float z  = ic[0];
    float th = thrp[0];
    float e = 0.f, c = 0.f;
    for (int k = 0; k < TRAJ_LEN; ++k) {
      float cur = z;                       // scan emits z BEFORE update
      traj[k] = cur;
      e += cur * cur;
      c += (cur > th) ? 1.f : 0.f;
      prefE[k] = e;
      prefC[k] = c;
      z = (cur < th) ? (cur / th) : ((1.f - cur) / (1.f - th));
    }
  }
}

// ---- Kernel 2: per-element first-hit search + feature computation ---------
__global__ void chaos_feat_kernel(const float* __restrict__ x,
                                  const float* __restrict__ epsp,
                                  const float* __restrict__ traj,
                                  const float* __restrict__ prefE,
                                  const float* __restrict__ prefC,
                                  float* __restrict__ feats) {
  __shared__ float s_traj[CHUNK];
  __shared__ int   s_alive;

  const int   idx = blockIdx.x * blockDim.x + threadIdx.x;   // < 65536 exactly
  const float xv  = x[idx];
  const float eps = epsp[0];

  int kFire = -1;
  for (int base = 0; base < TRAJ_LEN; base += CHUNK) {
    const int cn = (TRAJ_LEN - base < CHUNK) ? (TRAJ_LEN - base) : CHUNK;
    // cooperative LDS stage of this trajectory chunk
    for (int i = threadIdx.x; i < cn; i += blockDim.x)
      s_traj[i] = traj[base + i];
    if (threadIdx.x == 0) s_alive = 0;
    // pull the next chunk toward the WGP while we scan this one
    if (base + CHUNK < TRAJ_LEN)
      __builtin_prefetch(traj + base + CHUNK + threadIdx.x * 8, 0, 0);
    __syncthreads();                       // (1) chunk + reset visible

    if (kFire < 0) {
      for (int i = 0; i < cn; ++i) {
        if (fabsf(s_traj[i] - xv) < eps) { kFire = base + i; break; }
      }
      if (kFire < 0) atomicOr(&s_alive, 1);
    }
    __syncthreads();                       // (2) all votes in
    const int alive = s_alive;
    __syncthreads();                       // (3) all reads done before reuse
    if (!alive) break;                     // whole block fired -> early exit
  }

  const int   k      = (kFire < 0) ? (TRAJ_LEN - 1) : kFire;
  const float tt     = (kFire < 0) ? (float)TRAJ_LEN : (float)(kFire + 1);
  const float energy = prefE[k];
  const float cnt    = prefC[k];
  const float p      = cnt / tt;
  const float q      = 1.f - p;
  const float e0     = (p > 0.f) ? p * log2f(p) : 0.f;
  const float e1     = (q > 0.f) ? q * log2f(q) : 0.f;
  const float ent    = -(e0 + e1);

  // feats laid out [N, F*4] row-major; element idx owns 4 contiguous floats
  float4 f4 = make_float4(tt, energy, p, ent);
  *reinterpret_cast<float4*>(feats + (size_t)idx * 4) = f4;
}

// ---- Kernel 3: out[2048,2] = feats[2048,128] @ W^T + b via fp32 WMMA ------
// One wave (32 lanes) per 16-row tile; K=128 as 32 chained 16x16x4 WMMAs.
// B tile (K x 16, cols 2..15 zero) staged in LDS to keep the WMMA loop free
// of exec-mask divergence.
__global__ void chaos_gemm_kernel(const float* __restrict__ feats,
                                  const float* __restrict__ W,
                                  const float* __restrict__ b,
                                  float* __restrict__ out) {
  __shared__ float s_B[16 * 128];      // [n][k], zero padded, 8 KB

  const int lane = threadIdx.x;        // 0..31
  const int tile = blockIdx.x;         // 0..127
  const int m    = lane & 15;
  const int hi   = lane >> 4;          // 0 or 1

  // Fill padded B: unconditional load from clamped row, cndmask to zero.
  for (int i = lane; i < 16 * 128; i += 32) {
    const int n = i >> 7;
    const int k = i & 127;
    const float v = W[(n & 1) * 128 + k];     // rows 0/1 only, always in-bounds
    s_B[i] = (n < 2) ? v : 0.f;
  }
  __syncthreads();

  const float* __restrict__ arow = feats + (size_t)(tile * 16 + m) * 128;
  const float* __restrict__ brow = s_B + m * 128;

  v8f c = {};
  #pragma unroll 4
  for (int k0 = 0; k0 < 128; k0 += 4) {
    // A (16x4 f32): lanes 0-15 hold M=lane, K={k0,k0+1}; lanes 16-31 K={k0+2,k0+3}
    v2f a;
    a.x = arow[k0 + hi * 2 + 0];
    a.y = arow[k0 + hi * 2 + 1];
    // B (4x16 f32): VGPR0 = row K=k0 (lanes 0-15) / K=k0+2 (lanes 16-31); VGPR1 = +1.
    v2f bb;
    bb.x = brow[k0 + hi * 2 + 0];
    bb.y = brow[k0 + hi * 2 + 1];
    c = __builtin_amdgcn_wmma_f32_16x16x4_f32(
        /*neg_a=*/false, a, /*neg_b=*/false, bb,
        /*c_mod=*/(short)0, c, /*reuse_a=*/false, /*reuse_b=*/false);
  }

  // D (16x16 f32): VGPR r -> M = r + hi*8, N = m. Only N<2 is real output.
  if (m < 2) {
    const float bias = b[m];
    #pragma unroll
    for (int r = 0; r < 8; ++r) {
      const int row = tile * 16 + hi * 8 + r;
      out[row * 2 + m] = c[r] + bias;
    }
  }
}

// ---------------------------------------------------------------------------
extern "C" void kernel_launch(void* const* d_in, const int* in_sizes, int n_in,
                              void* d_out, int out_size, void* d_ws, size_t ws_size,
                              hipStream_t stream) {
  const float* x   = (const float*)d_in[0];  // [2048,32]
  const float* ic  = (const float*)d_in[1];  // [1]
  const float* thr = (const float*)d_in[2];  // [1]
  const float* eps = (const float*)d_in[3];  // [1]
  const float* W   = (const float*)d_in[4];  // [2,128]
  const float* b   = (const float*)d_in[5];  // [2]
  float* out = (float*)d_out;                // [2048,2]

  // workspace layout (floats): traj | prefE | prefC | feats
  float* ws    = (float*)d_ws;
  float* traj  = ws;                 // 10240 (padded)
  float* prefE = ws + 10240;         // 10240
  float* prefC = ws + 20480;         // 10240
  float* feats = ws + 30720;         // 2048*128

  chaos_traj_kernel<<<1, 32, 0, stream>>>(ic, thr, traj, prefE, prefC);
  chaos_feat_kernel<<<NELEM / 256, 256, 0, stream>>>(x, eps, traj, prefE, prefC, feats);
  chaos_gemm_kernel<<<2048 / 16, 32, 0, stream>>>(feats, W, b, out);
}